// SVDPlusPlus_84361747628058
// MI455X (gfx1250) — compile-verified
//
#include <hip/hip_runtime.h>

#define NUM_F      64
#define MAX_HIST   200
#define MU_CONST   3.5f

#define WAVES_PER_BLOCK 16
#define BLOCK_THREADS   (WAVES_PER_BLOCK * 32)
#define LDS_STRIDE      68   // 64 + pad to rotate banks row-to-row

typedef float v2f __attribute__((ext_vector_type(2)));
typedef float v8f __attribute__((ext_vector_type(8)));

__global__ __launch_bounds__(BLOCK_THREADS)
void svdpp_score_kernel(const int* __restrict__ user,
                        const int* __restrict__ item,
                        const int* __restrict__ implicit,
                        const int* __restrict__ hist_len,
                        const float* __restrict__ P,
                        const float* __restrict__ Q,
                        const float* __restrict__ Y,
                        const float* __restrict__ bu,
                        const float* __restrict__ bi,
                        float* __restrict__ out,
                        int batch)
{
    __shared__ float Pt[WAVES_PER_BLOCK][LDS_STRIDE];  // P rows, 16 x 64
    __shared__ float Zt[WAVES_PER_BLOCK][LDS_STRIDE];  // Z = Q + norm*y_sum rows
    __shared__ float Dt[16][17];                       // WMMA D tile scatter
    __shared__ float Bui[WAVES_PER_BLOCK];

    const int tid  = threadIdx.x;
    const int wave = tid >> 5;
    const int lane = tid & 31;
    const int b    = blockIdx.x * WAVES_PER_BLOCK + wave;

    // ---------------- Phase 1: one wave32 per batch element ----------------
    // Lane owns a float2 slice of F (lane*2, lane*2+1): one global_load_b64
    // per wave fetches exactly one 256B embedding row.
    if (b < batch) {
        const int u  = user[b];
        const int it = item[b];
        int len = hist_len[b];
        if (len < 1) len = 1;
        if (len > MAX_HIST) len = MAX_HIST;
        const float norm = 1.0f / sqrtf((float)len);

        const float2 p = *(const float2*)(P + (size_t)u  * NUM_F + lane * 2);
        const float2 q = *(const float2*)(Q + (size_t)it * NUM_F + lane * 2);

        float ax = 0.0f, ay = 0.0f;
        const int* __restrict__ imp = implicit + (size_t)b * MAX_HIST;

        for (int h0 = 0; h0 < len; h0 += 32) {
            const int hh    = h0 + lane;
            const int idx_l = (hh < len) ? imp[hh] : 0;   // 32 indices per trip
            const int cnt   = min(32, len - h0);
            int j = 0;
            // 4 gathers in flight per wave (pipelined b64 loads)
            for (; j + 4 <= cnt; j += 4) {
                const int i0 = __shfl(idx_l, j + 0, 32);
                const int i1 = __shfl(idx_l, j + 1, 32);
                const int i2 = __shfl(idx_l, j + 2, 32);
                const int i3 = __shfl(idx_l, j + 3, 32);
                const float2 y0 = *(const float2*)(Y + (size_t)i0 * NUM_F + lane * 2);
                const float2 y1 = *(const float2*)(Y + (size_t)i1 * NUM_F + lane * 2);
                const float2 y2 = *(const float2*)(Y + (size_t)i2 * NUM_F + lane * 2);
                const float2 y3 = *(const float2*)(Y + (size_t)i3 * NUM_F + lane * 2);
                ax += y0.x; ay += y0.y;
                ax += y1.x; ay += y1.y;
                ax += y2.x; ay += y2.y;
                ax += y3.x; ay += y3.y;
            }
            for (; j < cnt; ++j) {
                const int i0 = __shfl(idx_l, j, 32);
                const float2 y0 = *(const float2*)(Y + (size_t)i0 * NUM_F + lane * 2);
                ax += y0.x; ay += y0.y;
            }
        }

        // Z = Q_i + norm * y_sum  (the vector P_u is dotted with)
        Pt[wave][lane * 2]     = p.x;
        Pt[wave][lane * 2 + 1] = p.y;
        Zt[wave][lane * 2]     = q.x + norm * ax;
        Zt[wave][lane * 2 + 1] = q.y + norm * ay;
        if (lane == 0) Bui[wave] = MU_CONST + bu[u] + bi[it];
    }
    __syncthreads();

    // ---------------- Phase 2: 16 dot products on the matrix pipe ----------
    // D(16x16) = A(16x4) x B(4x16) chained over K=64 in 16 WMMA steps.
    // A[m,k] = Pt[m][k], B[k,n] = Zt[n][k]  =>  D[m,m] = dot(P_m, Z_m).
    if (wave == 0) {
#if __has_builtin(__builtin_amdgcn_wmma_f32_16x16x4_f32)
        const int row  = lane & 15;
        const int koff = (lane >> 4) << 1;   // lanes 0-15: K, K+1 ; lanes 16-31: K+2, K+3
        v8f c = {};
        #pragma unroll
        for (int k0 = 0; k0 < NUM_F; k0 += 4) {
            v2f a, bm;
            a.x  = Pt[row][k0 + koff];
            a.y  = Pt[row][k0 + koff + 1];
            bm.x = Zt[row][k0 + koff];
            bm.y = Zt[row][k0 + koff + 1];
            c = __builtin_amdgcn_wmma_f32_16x16x4_f32(
                    /*neg_a=*/false, a, /*neg_b=*/false, bm,
                    /*c_mod=*/(short)0, c, /*reuse_a=*/false, /*reuse_b=*/false);
        }
        // C/D layout: VGPR r, lane l -> M = r + 8*(l>=16), N = l&15
        {
            const int mbase = (lane >> 4) << 3;
            #pragma unroll
            for (int r = 0; r < 8; ++r)
                Dt[mbase + r][lane & 15] = c[r];
        }
#else
        // Fallback: scalar dots (should not be needed on gfx1250)
        if (lane < 16) {
            float s = 0.0f;
            for (int f = 0; f < NUM_F; ++f) s += Pt[lane][f] * Zt[lane][f];
            Dt[lane][lane] = s;
        }
#endif
    }
    __syncthreads();

    if (wave == 0 && lane < 16) {
        const int ob = blockIdx.x * WAVES_PER_BLOCK + lane;
        if (ob < batch) out[ob] = Bui[lane] + Dt[lane][lane];
    }
}

extern "C" void kernel_launch(void* const* d_in, const int* in_sizes, int n_in,
                              void* d_out, int out_size, void* d_ws, size_t ws_size,
                              hipStream_t stream) {
    const int*   user     = (const int*)d_in[0];
    const int*   item     = (const int*)d_in[1];
    const int*   implicit = (const int*)d_in[2];
    const int*   hist_len = (const int*)d_in[3];
    const float* P        = (const float*)d_in[4];
    const float* Q        = (const float*)d_in[5];
    const float* Y        = (const float*)d_in[6];
    const float* bu       = (const float*)d_in[7];
    const float* bi       = (const float*)d_in[8];
    float*       out      = (float*)d_out;

    const int batch  = in_sizes[0];
    const int blocks = (batch + WAVES_PER_BLOCK - 1) / WAVES_PER_BLOCK;

    svdpp_score_kernel<<<blocks, BLOCK_THREADS, 0, stream>>>(
        user, item, implicit, hist_len, P, Q, Y, bu, bi, out, batch);
}